// GCN_single_31327491457680
// MI455X (gfx1250) — compile-verified
//
#include <hip/hip_runtime.h>
#include <math.h>

#define N_NODES 50000
#define N_EDGES 640000
#define DIM     128

typedef __attribute__((ext_vector_type(2))) float v2f;
typedef __attribute__((ext_vector_type(8))) float v8f;

// ---------------- degree / normalization ----------------

__global__ void k_deg_init(unsigned* __restrict__ deg) {
    int i = blockIdx.x * blockDim.x + threadIdx.x;
    if (i < N_NODES) deg[i] = 1u;              // self-loop
}

__global__ void k_deg_count(const int* __restrict__ dst, unsigned* __restrict__ deg) {
    int e = blockIdx.x * blockDim.x + threadIdx.x;
    if (e < N_EDGES) atomicAdd(&deg[dst[e]], 1u);
}

__global__ void k_dinv(const unsigned* __restrict__ deg, float* __restrict__ dinv) {
    int i = blockIdx.x * blockDim.x + threadIdx.x;
    if (i < N_NODES) dinv[i] = rsqrtf((float)deg[i]);
}

// ---------------- fp32 WMMA GEMM: out[N,128] = A[N,128] @ W[128,128] (+bias) ----------------
// One wave -> one 16x16 output tile.  8 waves/block cover all 8 column tiles of
// one 16-row strip.  N_NODES % 16 == 0, so no bounds checks and EXEC stays all-1s.

__global__ __launch_bounds__(256) void k_gemm(const float* __restrict__ A,
                                              const float* __restrict__ W,
                                              const float* __restrict__ bias,
                                              float* __restrict__ out,
                                              int addBias) {
    const int wave = threadIdx.x >> 5;
    const int lane = threadIdx.x & 31;
    const int l15  = lane & 15;
    const int hi   = lane >> 4;          // lane half selects K pair
    const int rowBase = blockIdx.x << 4;
    const int colBase = wave << 4;

    // A: 16x4 tile. lanes 0-15: M=lane, K={k,k+1}; lanes 16-31: M=lane-16, K={k+2,k+3}
    const float* arow = A + (size_t)(rowBase + l15) * DIM + (hi << 1);
    // B: 4x16 tile. lane half selects the K pair, N = l15
    const float* wcol = W + (size_t)(hi << 1) * DIM + colBase + l15;

    v8f c = {};
#pragma unroll
    for (int k = 0; k < DIM; k += 4) {
        v2f a = *(const v2f*)(arow + k);          // contiguous, 8B aligned -> b64 load
        v2f b;
        b.x = wcol[(size_t)k * DIM];
        b.y = wcol[(size_t)(k + 1) * DIM];
        c = __builtin_amdgcn_wmma_f32_16x16x4_f32(false, a, false, b,
                                                  (short)0, c, false, false);
    }

    const float bv = addBias ? bias[colBase + l15] : 0.0f;
    const int m0 = hi << 3;                       // VGPR r -> M = r + 8*hi
    float* orow = out + (size_t)(rowBase + m0) * DIM + colBase + l15;
#pragma unroll
    for (int r = 0; r < 8; ++r)
        orow[(size_t)r * DIM] = c[r] + bv;
}

// ---------------- self-loop init: agg[i] = dinv[i]^2 * h[i] ----------------

__global__ __launch_bounds__(256) void k_self(const float* __restrict__ h,
                                              const float* __restrict__ dinv,
                                              float* __restrict__ agg) {
    int t = blockIdx.x * blockDim.x + threadIdx.x;   // float4 index
    int node = t >> 5;                               // 32 float4 per row
    float s = dinv[node];
    s *= s;
    float4 v = ((const float4*)h)[t];
    float4 o;
    o.x = v.x * s; o.y = v.y * s; o.z = v.z * s; o.w = v.w * s;
    ((float4*)agg)[t] = o;
}

// ---------------- edge scatter-add: agg[dst] += dinv[src]*dinv[dst] * h[src] ----------------
// One edge per wave; each lane owns 4 contiguous features.  L2-resident atomics.

__global__ __launch_bounds__(256) void k_scatter(const int* __restrict__ src,
                                                 const int* __restrict__ dst,
                                                 const float* __restrict__ dinv,
                                                 const float* __restrict__ h,
                                                 float* __restrict__ agg) {
    int w    = (blockIdx.x * blockDim.x + threadIdx.x) >> 5;  // edge id
    int lane = threadIdx.x & 31;
    int s = src[w];
    int d = dst[w];
    float norm = dinv[s] * dinv[d];
    float4 v = ((const float4*)(h + (size_t)s * DIM))[lane];
    float* ap = agg + (size_t)d * DIM + (lane << 2);
    unsafeAtomicAdd(ap + 0, norm * v.x);
    unsafeAtomicAdd(ap + 1, norm * v.y);
    unsafeAtomicAdd(ap + 2, norm * v.z);
    unsafeAtomicAdd(ap + 3, norm * v.w);
}

// ---------------- activation: h = tanh(agg + b1) ----------------

__global__ __launch_bounds__(256) void k_act(const float* __restrict__ agg,
                                             const float* __restrict__ b1,
                                             float* __restrict__ h) {
    int t = blockIdx.x * blockDim.x + threadIdx.x;
    h[t] = tanhf(agg[t] + b1[t & (DIM - 1)]);
}

// ---------------- host launch ----------------

extern "C" void kernel_launch(void* const* d_in, const int* in_sizes, int n_in,
                              void* d_out, int out_size, void* d_ws, size_t ws_size,
                              hipStream_t stream) {
    const float* x  = (const float*)d_in[0];
    const int*   ei = (const int*)  d_in[1];
    const float* W1 = (const float*)d_in[2];
    const float* b1 = (const float*)d_in[3];
    const float* W2 = (const float*)d_in[4];
    const float* b2 = (const float*)d_in[5];
    float* out = (float*)d_out;

    float*    ws   = (float*)d_ws;
    unsigned* deg  = (unsigned*)ws;                       // N_NODES u32
    float*    dinv = ws + 65536;                          // N_NODES f32
    float*    h    = ws + 131072;                         // N*128 f32
    float*    agg  = ws + 131072 + (size_t)N_NODES * DIM; // N*128 f32

    const int* esrc = ei;             // edge_index[0]
    const int* edst = ei + N_EDGES;   // edge_index[1]

    k_deg_init <<<(N_NODES + 255) / 256, 256, 0, stream>>>(deg);
    k_deg_count<<<(N_EDGES + 255) / 256, 256, 0, stream>>>(edst, deg);
    k_dinv     <<<(N_NODES + 255) / 256, 256, 0, stream>>>(deg, dinv);

    k_gemm <<<N_NODES / 16, 256, 0, stream>>>(x, W1, b1, h, 0);        // h = x @ W1
    k_self <<<(N_NODES * DIM / 4) / 256, 256, 0, stream>>>(h, dinv, agg);
    k_scatter<<<N_EDGES / 8, 256, 0, stream>>>(esrc, edst, dinv, h, agg);
    k_act  <<<(N_NODES * DIM) / 256, 256, 0, stream>>>(agg, b1, h);    // h = tanh(agg+b1)
    k_gemm <<<N_NODES / 16, 256, 0, stream>>>(h, W2, b2, out, 1);      // out = h @ W2 + b2
}